// GraphAttentionLayer_69466801045804
// MI455X (gfx1250) — compile-verified
//
#include <hip/hip_runtime.h>
#include <math.h>

// ---------------------------------------------------------------------------
// GAT layer for MI455X (gfx1250, wave32).
//   h = x @ W                      -> v_wmma_f32_16x16x4_f32 (exact f32 MMA)
//   a_src/a_dst per-node logits    -> thread-per-node
//   segment softmax over dst       -> atomicMax / atomicAdd passes (L2 atomics)
//   out[dst] += alpha * h[src]     -> wave-per-edge coalesced atomic adds
// ---------------------------------------------------------------------------

typedef __attribute__((ext_vector_type(2))) float v2f;
typedef __attribute__((ext_vector_type(8))) float v8f;

#define IN_CH     128
#define OUT_CH    64          // H * OUT = 2 * 32
#define NEG_SLOPE 0.2f

// Float atomic max via int punning (works for all finite values when the
// location is initialized to -inf).
__device__ __forceinline__ void atomic_max_f32(float* addr, float val) {
  int vi = __float_as_int(val);
  if (vi >= 0) {
    atomicMax((int*)addr, vi);
  } else {
    atomicMin((unsigned int*)addr, __float_as_uint(val));
  }
}

// ---------------------------------------------------------------------------
// Kernel 1: h[N,64] = x[N,128] @ W[128,64] via V_WMMA_F32_16X16X4_F32.
// One wave per 16x16 output tile; 8 waves (256 threads) per block.
// W is staged in LDS once per block (32 KB of the 320 KB WGP LDS).
//
// Fragment layouts (ISA 7.12.2, wave32):
//   A 16x4 f32 : lanes 0-15 -> {K=0 (v0), K=1 (v1)}, lanes 16-31 -> {K=2, K=3}
//   B 4x16 f32 : v0 = row K=0 (lanes 0-15) / K=2 (lanes 16-31); v1 = K=1 / K=3
//   C 16x16 f32: VGPR g -> M = g (lanes 0-15) or g+8 (lanes 16-31), N = lane&15
// ---------------------------------------------------------------------------
__global__ __launch_bounds__(256) void gat_gemm_wmma(
    const float* __restrict__ x, const float* __restrict__ W,
    float* __restrict__ h, int n) {
  __shared__ float Wlds[IN_CH * OUT_CH];
  for (int i = threadIdx.x; i < IN_CH * OUT_CH; i += 256)
    Wlds[i] = W[i];
  __syncthreads();

  const int wave = threadIdx.x >> 5;
  const int lane = threadIdx.x & 31;
  const int half = lane >> 4;   // 0: lanes 0-15, 1: lanes 16-31
  const int m    = lane & 15;

  const int numRowTiles = (n + 15) >> 4;   // N=50000 -> 3125 exact tiles
  const int tile = blockIdx.x * 8 + wave;
  if (tile >= numRowTiles * 4) return;
  const int rowBase = (tile >> 2) * 16;
  const int colBase = (tile & 3) * 16;

  const int loadRow = min(rowBase + m, n - 1);  // clamp keeps EXEC full
  const float* __restrict__ xrow = x + (size_t)loadRow * IN_CH;

  v8f acc = {};
#pragma unroll
  for (int k0 = 0; k0 < IN_CH; k0 += 4) {
    const int ka = k0 + half * 2;
    v2f a;
    a.x = xrow[ka];
    a.y = xrow[ka + 1];
    v2f b;
    b.x = Wlds[ka * OUT_CH + colBase + m];
    b.y = Wlds[(ka + 1) * OUT_CH + colBase + m];
    acc = __builtin_amdgcn_wmma_f32_16x16x4_f32(
        /*neg_a=*/false, a, /*neg_b=*/false, b,
        /*c_mod=*/(short)0, acc, /*reuse_a=*/false, /*reuse_b=*/false);
  }

#pragma unroll
  for (int g = 0; g < 8; ++g) {
    const int row = rowBase + half * 8 + g;
    if (row < n) h[(size_t)row * OUT_CH + colBase + m] = acc[g];
  }
}

// ---------------------------------------------------------------------------
// Kernel 2: per-node attention logits + init of seg_max / denom.
// ---------------------------------------------------------------------------
__global__ __launch_bounds__(256) void gat_logits(
    const float* __restrict__ h, const float* __restrict__ att_src,
    const float* __restrict__ att_dst, float* __restrict__ a_src,
    float* __restrict__ a_dst, float* __restrict__ seg_max,
    float* __restrict__ denom, int n) {
  const int nid = blockIdx.x * 256 + threadIdx.x;
  if (nid >= n) return;
  const float* __restrict__ hr = h + (size_t)nid * OUT_CH;
#pragma unroll
  for (int hd = 0; hd < 2; ++hd) {
    float s = 0.f, d = 0.f;
#pragma unroll
    for (int c = 0; c < 32; ++c) {
      const float v = hr[hd * 32 + c];
      s = fmaf(v, att_src[hd * 32 + c], s);
      d = fmaf(v, att_dst[hd * 32 + c], d);
    }
    a_src[nid * 2 + hd]   = s;
    a_dst[nid * 2 + hd]   = d;
    seg_max[nid * 2 + hd] = -INFINITY;
    denom[nid * 2 + hd]   = 0.f;
  }
}

// ---------------------------------------------------------------------------
// Kernel 3: out rows <- bias (aggregation atomically accumulates on top).
// ---------------------------------------------------------------------------
__global__ __launch_bounds__(256) void gat_bias_init(
    float* __restrict__ out, const float* __restrict__ bias, int total) {
  const int idx = blockIdx.x * 256 + threadIdx.x;
  if (idx < total) out[idx] = bias[idx & (OUT_CH - 1)];
}

// ---------------------------------------------------------------------------
// Kernel 4: per-(edge,head) leaky-relu score -> segment atomic max over dst.
// Edge i in [0,E) comes from edge_index; i in [E, E+N) is the self-loop i-E.
// ---------------------------------------------------------------------------
__global__ __launch_bounds__(256) void gat_edge_max(
    const int* __restrict__ ei, const float* __restrict__ a_src,
    const float* __restrict__ a_dst, float* __restrict__ seg_max,
    int E, int n) {
  const int t = blockIdx.x * 256 + threadIdx.x;
  if (t >= (E + n) * 2) return;
  const int i = t >> 1, hd = t & 1;
  const int s = (i < E) ? ei[i]     : (i - E);
  const int d = (i < E) ? ei[E + i] : (i - E);
  float e = a_src[s * 2 + hd] + a_dst[d * 2 + hd];
  e = (e > 0.f) ? e : NEG_SLOPE * e;
  atomic_max_f32(&seg_max[d * 2 + hd], e);
}

// ---------------------------------------------------------------------------
// Kernel 5: per-(edge,head) exp(e - max) -> segment atomic sum (denominator).
// ---------------------------------------------------------------------------
__global__ __launch_bounds__(256) void gat_edge_sum(
    const int* __restrict__ ei, const float* __restrict__ a_src,
    const float* __restrict__ a_dst, const float* __restrict__ seg_max,
    float* __restrict__ denom, int E, int n) {
  const int t = blockIdx.x * 256 + threadIdx.x;
  if (t >= (E + n) * 2) return;
  const int i = t >> 1, hd = t & 1;
  const int s = (i < E) ? ei[i]     : (i - E);
  const int d = (i < E) ? ei[E + i] : (i - E);
  float e = a_src[s * 2 + hd] + a_dst[d * 2 + hd];
  e = (e > 0.f) ? e : NEG_SLOPE * e;
  atomicAdd(&denom[d * 2 + hd], __expf(e - seg_max[d * 2 + hd]));
}

// ---------------------------------------------------------------------------
// Kernel 6: wave32 per edge. Lane j handles channel j (head 0) and j+32
// (head 1): coalesced gather of h[src], coalesced global_atomic_add_f32 into
// out[dst]. Alphas recomputed per lane (cheap scalar math, avoids 13 MB of
// per-edge scratch).
// ---------------------------------------------------------------------------
__global__ __launch_bounds__(256) void gat_edge_aggr(
    const int* __restrict__ ei, const float* __restrict__ a_src,
    const float* __restrict__ a_dst, const float* __restrict__ seg_max,
    const float* __restrict__ denom, const float* __restrict__ h,
    float* __restrict__ out, int E, int n) {
  const int wid  = (blockIdx.x * 256 + threadIdx.x) >> 5;
  const int lane = threadIdx.x & 31;
  if (wid >= E + n) return;
  const int s = (wid < E) ? ei[wid]     : (wid - E);
  const int d = (wid < E) ? ei[E + wid] : (wid - E);

  float e0 = a_src[s * 2] + a_dst[d * 2];
  e0 = (e0 > 0.f) ? e0 : NEG_SLOPE * e0;
  const float al0 = __expf(e0 - seg_max[d * 2]) / denom[d * 2];

  float e1 = a_src[s * 2 + 1] + a_dst[d * 2 + 1];
  e1 = (e1 > 0.f) ? e1 : NEG_SLOPE * e1;
  const float al1 = __expf(e1 - seg_max[d * 2 + 1]) / denom[d * 2 + 1];

  const float* __restrict__ hs = h + (size_t)s * OUT_CH;
  float* __restrict__ od = out + (size_t)d * OUT_CH;
  atomicAdd(&od[lane],      al0 * hs[lane]);
  atomicAdd(&od[lane + 32], al1 * hs[lane + 32]);
}

// ---------------------------------------------------------------------------
// Launch. Inputs (setup_inputs order): x[N,128] f32, edge_index[2,E] int,
// W[128,64] f32, att_src[2,32] f32, att_dst[2,32] f32, bias[64] f32.
// Output: [N, 64] f32.
// ---------------------------------------------------------------------------
extern "C" void kernel_launch(void* const* d_in, const int* in_sizes, int n_in,
                              void* d_out, int out_size, void* d_ws,
                              size_t ws_size, hipStream_t stream) {
  const float* x       = (const float*)d_in[0];
  const int*   ei      = (const int*)d_in[1];
  const float* W       = (const float*)d_in[2];
  const float* att_src = (const float*)d_in[3];
  const float* att_dst = (const float*)d_in[4];
  const float* bias    = (const float*)d_in[5];
  float*       out     = (float*)d_out;

  const int n = in_sizes[0] / IN_CH;   // 50000
  const int E = in_sizes[1] / 2;       // 1,600,000

  // Workspace layout (floats): h[N*64] | a_src[N*2] | a_dst[N*2] |
  //                            seg_max[N*2] | denom[N*2]   (~14.5 MB)
  float* ws     = (float*)d_ws;
  float* h      = ws;
  float* a_src  = h + (size_t)n * OUT_CH;
  float* a_dst  = a_src + (size_t)n * 2;
  float* smax   = a_dst + (size_t)n * 2;
  float* den    = smax + (size_t)n * 2;

  const int rowTiles = (n + 15) / 16;
  const int tiles    = rowTiles * 4;

  gat_gemm_wmma<<<(tiles + 7) / 8, 256, 0, stream>>>(x, W, h, n);
  gat_logits<<<(n + 255) / 256, 256, 0, stream>>>(h, att_src, att_dst, a_src,
                                                  a_dst, smax, den, n);
  gat_bias_init<<<(n * OUT_CH + 255) / 256, 256, 0, stream>>>(out, bias,
                                                              n * OUT_CH);
  const int tot2 = (E + n) * 2;
  gat_edge_max<<<(tot2 + 255) / 256, 256, 0, stream>>>(ei, a_src, a_dst, smax,
                                                       E, n);
  gat_edge_sum<<<(tot2 + 255) / 256, 256, 0, stream>>>(ei, a_src, a_dst, smax,
                                                       den, E, n);
  gat_edge_aggr<<<(E + n + 7) / 8, 256, 0, stream>>>(ei, a_src, a_dst, smax,
                                                     den, h, out, E, n);
}